// MoDRouter_62612033241432
// MI455X (gfx1250) — compile-verified
//
#include <hip/hip_runtime.h>
#include <math.h>

typedef __attribute__((ext_vector_type(2))) float v2f;
typedef __attribute__((ext_vector_type(8))) float v8f;

#define BB 4
#define TT 8192
#define DD 2048
#define KK 4096

// ---------------------------------------------------------------------------
// Kernel 1: scores[b,t] = dot(x[b,t,:], W) + entropy[b,t] via V_WMMA_F32_16X16X4_F32
// One wave handles 16 flattened rows. B-matrix columns all carry the same W
// chunk, so every column of the 16x16 accumulator holds the row's dot product.
// Four independent accumulators break the WMMA->WMMA RAW chain so the four
// wmma ops per iteration issue back-to-back and hide under the b64 loads.
// ---------------------------------------------------------------------------
__global__ __launch_bounds__(256) void mod_scores_kernel(
    const float* __restrict__ x,        // (B*T, D)
    const float* __restrict__ entropy,  // (B*T)
    const float* __restrict__ W,        // (D)
    float* __restrict__ scores,         // (B*T)
    int D, int nRows)
{
    const int lane = threadIdx.x & 31;
    const int waveInBlk = threadIdx.x >> 5;
    const int wave = blockIdx.x * (blockDim.x >> 5) + waveInBlk;
    const int rowBase = wave * 16;
    if (rowBase >= nRows) return;   // wave-uniform: EXEC stays all-ones for WMMA

    const int half = lane >> 4;     // 0: K=0..1, 1: K=2..3 (per 16x4 f32 A layout)
    const int m = lane & 15;        // row within tile

    const float* __restrict__ xrow = x + (size_t)(rowBase + m) * (size_t)D + half * 2;
    const float* __restrict__ wrow = W + half * 2;

    v8f c0 = {}, c1 = {}, c2 = {}, c3 = {};
    for (int kc = 0; kc < D; kc += 16) {
        v2f a0 = *reinterpret_cast<const v2f*>(xrow + kc);       // global_load_b64
        v2f b0 = *reinterpret_cast<const v2f*>(wrow + kc);
        v2f a1 = *reinterpret_cast<const v2f*>(xrow + kc + 4);
        v2f b1 = *reinterpret_cast<const v2f*>(wrow + kc + 4);
        v2f a2 = *reinterpret_cast<const v2f*>(xrow + kc + 8);
        v2f b2 = *reinterpret_cast<const v2f*>(wrow + kc + 8);
        v2f a3 = *reinterpret_cast<const v2f*>(xrow + kc + 12);
        v2f b3 = *reinterpret_cast<const v2f*>(wrow + kc + 12);
        // 8 args: neg_a, A, neg_b, B, c_mod, C, reuse_a, reuse_b
        c0 = __builtin_amdgcn_wmma_f32_16x16x4_f32(false, a0, false, b0, (short)0, c0, false, false);
        c1 = __builtin_amdgcn_wmma_f32_16x16x4_f32(false, a1, false, b1, (short)0, c1, false, false);
        c2 = __builtin_amdgcn_wmma_f32_16x16x4_f32(false, a2, false, b2, (short)0, c2, false, false);
        c3 = __builtin_amdgcn_wmma_f32_16x16x4_f32(false, a3, false, b3, (short)0, c3, false, false);
    }
    v8f c = (c0 + c1) + (c2 + c3);

    // C/D layout: lanes 0-15 hold M=0..7 in c[0..7]; lanes 16-31 hold M=8..15.
    // Column N is redundant -> lane with N==0 of each half writes its 8 rows.
    if (m == 0) {
        #pragma unroll
        for (int r = 0; r < 8; ++r) {
            const int row = rowBase + half * 8 + r;
            scores[row] = c[r] + entropy[row];
        }
    }
}

// ---------------------------------------------------------------------------
// Kernel 2: per-batch exact top-K via 4x8-bit radix select on monotonic keys,
// then two block scans to emit ascending-sorted positions. Ties at the
// threshold are taken in ascending index order (matches lax.top_k semantics).
// ---------------------------------------------------------------------------
__device__ __forceinline__ int block_exscan_1024(int v, volatile int* sbuf)
{
    const int tid = threadIdx.x;
    sbuf[tid] = v;
    __syncthreads();
    #pragma unroll
    for (int off = 1; off < 1024; off <<= 1) {
        int t = (tid >= off) ? sbuf[tid - off] : 0;
        __syncthreads();
        sbuf[tid] += t;
        __syncthreads();
    }
    int incl = sbuf[tid];
    __syncthreads();   // protect sbuf before any subsequent reuse
    return incl - v;
}

__global__ __launch_bounds__(1024) void mod_select_kernel(
    const float* __restrict__ scores,  // (B*T)
    float* __restrict__ idx_out,       // (B*K) written as float
    int* __restrict__ pos_of,          // (B*T) workspace: position or -1
    float* __restrict__ w_of,          // (B*T) workspace: sigmoid(score) if selected
    int T, int K)
{
    __shared__ unsigned keys[TT];      // 32 KB
    __shared__ int hist[256];
    __shared__ int sbuf[1024];
    __shared__ unsigned s_prefix;
    __shared__ int s_remK;

    const int tid = threadIdx.x;
    const int b = blockIdx.x;
    const int base = b * T;
    const int EPT = TT / 1024;         // 8 elements per thread

    // Load + monotonic-ascending key transform
    unsigned mykeys[EPT];
    #pragma unroll
    for (int e = 0; e < EPT; ++e) {
        const int t = tid * EPT + e;
        const float f = scores[base + t];
        unsigned u = __float_as_uint(f);
        u = (u & 0x80000000u) ? ~u : (u | 0x80000000u);
        mykeys[e] = u;
        keys[t] = u;
    }
    __syncthreads();

    // Radix select the K-th largest key
    unsigned prefix = 0;
    int remK = K;
    for (int shift = 24; shift >= 0; shift -= 8) {
        if (tid < 256) hist[tid] = 0;
        __syncthreads();
        const unsigned maskHigh = (shift == 24) ? 0u : (0xFFFFFFFFu << (shift + 8));
        #pragma unroll
        for (int e = 0; e < EPT; ++e) {
            const unsigned key = mykeys[e];
            if ((key & maskHigh) == (prefix & maskHigh))
                atomicAdd(&hist[(key >> shift) & 255], 1);
        }
        __syncthreads();
        if (tid == 0) {
            int acc = 0;
            int bin = 255;
            for (; bin > 0; --bin) {
                const int h = hist[bin];
                if (acc + h >= remK) break;
                acc += h;
            }
            s_prefix = prefix | ((unsigned)bin << shift);
            s_remK = remK - acc;
        }
        __syncthreads();
        prefix = s_prefix;
        remK = s_remK;
        __syncthreads();
    }
    const unsigned tau = prefix;   // threshold key
    const int need = remK;         // # of tau-valued elements to take (lowest idx first)

    // eq-rank scan
    int loc_eq = 0;
    #pragma unroll
    for (int e = 0; e < EPT; ++e) loc_eq += (mykeys[e] == tau) ? 1 : 0;
    const int ex_eq = block_exscan_1024(loc_eq, sbuf);

    // selection flags
    int sel_bits = 0, loc_sel = 0, eqr = ex_eq;
    #pragma unroll
    for (int e = 0; e < EPT; ++e) {
        bool s;
        const unsigned key = mykeys[e];
        if (key > tau)      s = true;
        else if (key == tau) { s = (eqr < need); ++eqr; }
        else                 s = false;
        if (s) { sel_bits |= (1 << e); ++loc_sel; }
    }
    const int ex_sel = block_exscan_1024(loc_sel, sbuf);

    // emit ascending-ordered indices + maps
    int p = ex_sel;
    #pragma unroll
    for (int e = 0; e < EPT; ++e) {
        const int t = tid * EPT + e;
        if (sel_bits & (1 << e)) {
            idx_out[b * K + p] = (float)t;
            pos_of[base + t] = p;
            const float s = scores[base + t];
            w_of[base + t] = 1.0f / (1.0f + expf(-s));
            ++p;
        } else {
            pos_of[base + t] = -1;
        }
    }
}

// ---------------------------------------------------------------------------
// Kernel 3: single streaming pass. One block per (b,t) row:
//   out[b,t,:] = x[b,t,:] (+ w * processed[b,p,:] if selected)
//   selected_x[b,p,:] = x[b,t,:] if selected
// ---------------------------------------------------------------------------
__global__ __launch_bounds__(256) void mod_fuse_kernel(
    const float* __restrict__ x,          // (B*T, D)
    const float* __restrict__ processed,  // (B, K, D)
    const int* __restrict__ pos_of,       // (B*T)
    const float* __restrict__ w_of,       // (B*T)
    float* __restrict__ selected_x,       // (B, K, D)
    float* __restrict__ out,              // (B*T, D)
    int T, int K, int D)
{
    const int r = blockIdx.x;             // flattened b*T + t
    const int b = r / T;
    const int p = pos_of[r];
    const int nd4 = D >> 2;

    const float4* __restrict__ xr = reinterpret_cast<const float4*>(x + (size_t)r * D);
    float4* __restrict__ outr = reinterpret_cast<float4*>(out + (size_t)r * D);

    if (p >= 0) {
        const float w = w_of[r];
        const size_t srow = ((size_t)b * K + (size_t)p) * (size_t)D;
        const float4* __restrict__ pr = reinterpret_cast<const float4*>(processed + srow);
        float4* __restrict__ sr = reinterpret_cast<float4*>(selected_x + srow);
        for (int i = threadIdx.x; i < nd4; i += blockDim.x) {
            const float4 xv = xr[i];   // global_load_b128
            const float4 pv = pr[i];
            sr[i] = xv;
            float4 ov;
            ov.x = fmaf(w, pv.x, xv.x);
            ov.y = fmaf(w, pv.y, xv.y);
            ov.z = fmaf(w, pv.z, xv.z);
            ov.w = fmaf(w, pv.w, xv.w);
            outr[i] = ov;
        }
    } else {
        for (int i = threadIdx.x; i < nd4; i += blockDim.x) {
            outr[i] = xr[i];
        }
    }
}

// ---------------------------------------------------------------------------
extern "C" void kernel_launch(void* const* d_in, const int* in_sizes, int n_in,
                              void* d_out, int out_size, void* d_ws, size_t ws_size,
                              hipStream_t stream)
{
    const float* x         = (const float*)d_in[0];  // (B,T,D)
    const float* entropy   = (const float*)d_in[1];  // (B,T)
    const float* processed = (const float*)d_in[2];  // (B,K,D)
    const float* W         = (const float*)d_in[3];  // (D)

    const int B = BB, T = TT, D = DD, K = KK;
    const int nRows = B * T;                         // 32768

    // d_out layout: selected_x | idx | scores | out (flat, return order)
    float* selected_x = (float*)d_out;
    float* idx_out    = selected_x + (size_t)B * K * D;
    float* scores     = idx_out + (size_t)B * K;
    float* outfull    = scores + (size_t)B * T;

    // workspace: pos_of (B*T int) then w_of (B*T float); pos_of fully rewritten each call, w_of only where selected
    int*   pos_of = (int*)d_ws;
    float* w_of   = (float*)((char*)d_ws + sizeof(int) * (size_t)(B * T));

    // 1) router scores via WMMA: 8 waves/block * 16 rows/wave = 128 rows/block
    const int rowsPerBlock = (256 / 32) * 16;
    mod_scores_kernel<<<nRows / rowsPerBlock, 256, 0, stream>>>(
        x, entropy, W, scores, D, nRows);

    // 2) exact top-K + ascending positions, one block per batch row
    mod_select_kernel<<<B, 1024, 0, stream>>>(scores, idx_out, pos_of, w_of, T, K);

    // 3) fused gather + weighted scatter-add streaming pass
    mod_fuse_kernel<<<nRows, 256, 0, stream>>>(
        x, processed, pos_of, w_of, selected_x, outfull, T, K, D);
}